// KLDivergenceLoss_44659069944161
// MI455X (gfx1250) — compile-verified
//
#include <hip/hip_runtime.h>
#include <stdint.h>

#define MD 6
#define NV 21           // 6*7/2 packed upper-triangular
#define TPB 256

#define AS1 __attribute__((address_space(1)))
#define AS3 __attribute__((address_space(3)))

typedef __attribute__((__vector_size__(4 * sizeof(int)))) int v4i;

#if __has_builtin(__builtin_amdgcn_global_load_async_to_lds_b128) && __has_builtin(__builtin_amdgcn_s_wait_asynccnt)
#define HAVE_ASYNC_LDS 1
#else
#define HAVE_ASYNC_LDS 0
#endif

// packed upper-triangular index, requires i<=j (row-major upper triangle, matches jnp.triu_indices order)
__device__ __forceinline__ constexpr int ut(int i, int j) {
    return i * MD - (i * (i - 1)) / 2 + (j - i);
}
// packed lower-triangular index, requires j<=i
__device__ __forceinline__ constexpr int lt(int i, int j) {
    return (i * (i + 1)) / 2 + j;
}
__device__ __forceinline__ float symv(const float* a, int i, int j) {
    return (i <= j) ? a[ut(i, j)] : a[ut(j, i)];
}

__device__ __forceinline__ float fast_rcp(float x) {
    float r = __builtin_amdgcn_rcpf(x);
    return r * (2.0f - x * r);   // one Newton step -> ~0.5 ulp
}

// Cholesky A = L L^T, then Ainv = L^-T L^-1 (packed upper). Returns prod(diag(L)) = sqrt(det A).
__device__ __forceinline__ float chol_inv6(const float (&A)[NV], float (&Ainv)[NV]) {
    float L[NV];
    float Dinv[MD];
    float prod = 1.0f;
#pragma unroll
    for (int j = 0; j < MD; ++j) {
        float s = A[ut(j, j)];
#pragma unroll
        for (int k = 0; k < MD; ++k)
            if (k < j) s -= L[lt(j, k)] * L[lt(j, k)];
        float d = __builtin_sqrtf(s);
        prod *= d;
        float r = fast_rcp(d);
        Dinv[j] = r;
        L[lt(j, j)] = d;
#pragma unroll
        for (int i = 0; i < MD; ++i)
            if (i > j) {
                float t = A[ut(j, i)];
#pragma unroll
                for (int k = 0; k < MD; ++k)
                    if (k < j) t -= L[lt(i, k)] * L[lt(j, k)];
                L[lt(i, j)] = t * r;
            }
    }
    // invert lower-triangular L into Li
    float Li[NV];
#pragma unroll
    for (int j = 0; j < MD; ++j) {
        Li[lt(j, j)] = Dinv[j];
#pragma unroll
        for (int i = 0; i < MD; ++i)
            if (i > j) {
                float s = 0.0f;
#pragma unroll
                for (int k = 0; k < MD; ++k)
                    if (k >= j && k < i) s += L[lt(i, k)] * Li[lt(k, j)];
                Li[lt(i, j)] = -s * Dinv[i];
            }
    }
    // Ainv(i,j) = sum_{k>=max(i,j)} Li(k,i)*Li(k,j), store i<=j packed upper
#pragma unroll
    for (int i = 0; i < MD; ++i)
#pragma unroll
        for (int j = 0; j < MD; ++j)
            if (j >= i) {
                float s = 0.0f;
#pragma unroll
                for (int k = 0; k < MD; ++k)
                    if (k >= j) s += Li[lt(k, i)] * Li[lt(k, j)];
                Ainv[ut(i, j)] = s;
            }
    return prod;
}

__global__ __launch_bounds__(TPB) void kl_partial_kernel(const float* __restrict__ pred,
                                                         const float* __restrict__ gt,
                                                         float* __restrict__ partial,
                                                         int nElem) {
    __shared__ __align__(16) float sp[TPB * NV];
    __shared__ __align__(16) float sg[TPB * NV];
    __shared__ float red[TPB];

    const int tid = threadIdx.x;
    const int blk = blockIdx.x;
    const long long base = (long long)blk * TPB * NV;       // dword offset of block chunk
    const long long totalDw = (long long)nElem * NV;
    const bool full = (long long)(blk + 1) * TPB <= (long long)nElem;

    if (full) {
        const float4* gp = (const float4*)(pred + base);
        const float4* gg = (const float4*)(gt + base);
        float4* lp = (float4*)sp;
        float4* lg = (float4*)sg;
        const int nV4 = (TPB * NV) / 4;  // 1344 float4 per matrix chunk
#if HAVE_ASYNC_LDS
        for (int k = tid; k < nV4; k += TPB) {
            __builtin_amdgcn_global_load_async_to_lds_b128(
                (AS1 v4i*)(uintptr_t)(gp + k),
                (AS3 v4i*)(uint32_t)(uintptr_t)(lp + k), 0, 0);
            __builtin_amdgcn_global_load_async_to_lds_b128(
                (AS1 v4i*)(uintptr_t)(gg + k),
                (AS3 v4i*)(uint32_t)(uintptr_t)(lg + k), 0, 0);
        }
        __builtin_amdgcn_s_wait_asynccnt(0);
#else
        for (int k = tid; k < nV4; k += TPB) {
            lp[k] = gp[k];
            lg[k] = gg[k];
        }
#endif
    } else {
        for (int k = tid; k < TPB * NV; k += TPB) {
            long long dw = base + k;
            sp[k] = (dw < totalDw) ? pred[dw] : 0.0f;
            sg[k] = (dw < totalDw) ? gt[dw] : 0.0f;
        }
    }
    __syncthreads();

    float kl = 0.0f;
    const long long idx = (long long)blk * TPB + tid;
    if (idx < (long long)nElem) {
        float P[NV], G[NV];
#pragma unroll
        for (int v = 0; v < NV; ++v) {
            P[v] = sp[tid * NV + v] + 1e-5f;   // scalar broadcast add in reference
            G[v] = sg[tid * NV + v];
        }
#pragma unroll
        for (int j = 0; j < MD; ++j) {
            P[ut(j, j)] += 1e-6f;
            G[ut(j, j)] += 1e-6f;
        }

        float Pi[NV], Gi[NV];
        const float prodp = chol_inv6(P, Pi);   // sqrt(det P)
        const float prodg = chol_inv6(G, Gi);   // sqrt(det G)

        const float log_det_ratio = 2.0f * __logf(prodp / prodg);

        // trace(Gi * P) with both symmetric
        float tr = 0.0f;
#pragma unroll
        for (int i = 0; i < MD; ++i)
#pragma unroll
            for (int j = 0; j < MD; ++j)
                tr += symv(Gi, i, j) * symv(P, i, j);

        // sum_{i,k,j} (Gi-Pi)[i,k] (P-G)[k,j] = sum_k rowsum(Gi-Pi)[k] * rowsum(P-G)[k]
        float diff = 0.0f;
#pragma unroll
        for (int k = 0; k < MD; ++k) {
            float rd = 0.0f, re = 0.0f;
#pragma unroll
            for (int j = 0; j < MD; ++j) {
                rd += symv(Gi, k, j) - symv(Pi, k, j);
                re += symv(P, k, j) - symv(G, k, j);
            }
            diff += rd * re;
        }

        kl = 0.5f * (log_det_ratio - (float)MD + tr + diff);
    }

    // deterministic block tree reduction
    red[tid] = kl;
    __syncthreads();
#pragma unroll
    for (int s = TPB / 2; s > 0; s >>= 1) {
        if (tid < s) red[tid] += red[tid + s];
        __syncthreads();
    }
    if (tid == 0) partial[blk] = red[0];
}

__global__ __launch_bounds__(TPB) void kl_reduce_kernel(const float* __restrict__ partial,
                                                        int nPart,
                                                        float* __restrict__ out,
                                                        float invN) {
    __shared__ float red[TPB];
    float s = 0.0f;
    for (int k = threadIdx.x; k < nPart; k += TPB) s += partial[k];
    red[threadIdx.x] = s;
    __syncthreads();
#pragma unroll
    for (int st = TPB / 2; st > 0; st >>= 1) {
        if (threadIdx.x < st) red[threadIdx.x] += red[threadIdx.x + st];
        __syncthreads();
    }
    if (threadIdx.x == 0) out[0] = red[0] * invN;
}

extern "C" void kernel_launch(void* const* d_in, const int* in_sizes, int n_in,
                              void* d_out, int out_size, void* d_ws, size_t ws_size,
                              hipStream_t stream) {
    const float* pred = (const float*)d_in[0];
    const float* gt   = (const float*)d_in[1];
    const int n = in_sizes[0] / NV;                 // batch elements
    const int blocks = (n + TPB - 1) / TPB;
    float* partial = (float*)d_ws;

    kl_partial_kernel<<<blocks, TPB, 0, stream>>>(pred, gt, partial, n);
    kl_reduce_kernel<<<1, TPB, 0, stream>>>(partial, blocks, (float*)d_out,
                                            1.0f / (float)n);
}